// GraphSage_16535624090440
// MI455X (gfx1250) — compile-verified
//
#include <hip/hip_runtime.h>
#include <hip/hip_bf16.h>

typedef __attribute__((ext_vector_type(2))) float v2f;
typedef __attribute__((ext_vector_type(8))) float v8f;

#define LDS_STRIDE 132  // 128 + 4 pad: rows land in distinct LDS banks

// ---------------------------------------------------------------------------
// Zero-fill scratch (agg + cnt) — grid-stride
// ---------------------------------------------------------------------------
__global__ void gs_zero_kernel(float* __restrict__ p, long n) {
  long i = (long)blockIdx.x * blockDim.x + threadIdx.x;
  long stride = (long)gridDim.x * blockDim.x;
  for (; i < n; i += stride) p[i] = 0.0f;
}

// ---------------------------------------------------------------------------
// Edge scatter: one wave (32 lanes x float4) per edge.
// agg[tgt] += h[src]  (128 fp32 atomics via 4 per lane), cnt[tgt] += 1.
// ---------------------------------------------------------------------------
__global__ void gs_scatter_kernel(const float* __restrict__ h,
                                  const int* __restrict__ src,
                                  const int* __restrict__ tgt,
                                  float* __restrict__ agg,
                                  float* __restrict__ cnt,
                                  int nEdges) {
  int lane = threadIdx.x & 31;
  int e = blockIdx.x * (blockDim.x >> 5) + (threadIdx.x >> 5);
  if (e >= nEdges) return;
  int s = src[e];
  int t = tgt[e];
  const float4 v = *(const float4*)(h + (long)s * 128 + lane * 4);
  float* dst = agg + (long)t * 128 + lane * 4;
  atomicAdd(dst + 0, v.x);
  atomicAdd(dst + 1, v.y);
  atomicAdd(dst + 2, v.z);
  atomicAdd(dst + 3, v.w);
  if (lane == 0) atomicAdd(cnt + t, 1.0f);
}

// ---------------------------------------------------------------------------
// SAGE layer GEMM via V_WMMA_F32_16X16X4_F32 (fp32 matrix pipe):
//   hout = act( mean @ Wl + b + hself @ Wr ),  mean = agg / max(cnt,1)
// Block: 256 threads = 8 wave32. Block computes 16 rows x 128 cols.
// Each wave owns one 16x16 tile; 64 chained WMMAs (K=128 twice).
// A/B/C fragment layouts per CDNA5 ISA 7.12.2 (16x4 f32 A, 8-VGPR C/D).
// ---------------------------------------------------------------------------
__global__ void __launch_bounds__(256)
gs_sage_wmma_kernel(const float* __restrict__ agg,
                    const float* __restrict__ cnt,
                    const float* __restrict__ hself,
                    const float* __restrict__ Wl,
                    const float* __restrict__ bl,
                    const float* __restrict__ Wr,
                    float* __restrict__ hout,
                    int doRelu, int nNodes) {
  __shared__ float lmean[16 * LDS_STRIDE];
  __shared__ float lself[16 * LDS_STRIDE];

  const int tid  = threadIdx.x;
  const int row0 = blockIdx.x * 16;

  // Cooperative stage: 16 rows x 128 cols of mean and self-features into LDS.
  for (int i = 0; i < 8; ++i) {
    int idx  = tid + i * 256;          // 0..2047
    int r    = idx >> 7;               // row within tile
    int c    = idx & 127;              // feature column
    int node = row0 + r;
    int nclamp = node < nNodes ? node : (nNodes - 1);
    float ct   = cnt[nclamp];
    float rinv = 1.0f / fmaxf(ct, 1.0f);
    lmean[r * LDS_STRIDE + c] = agg[(long)nclamp * 128 + c] * rinv;
    lself[r * LDS_STRIDE + c] = hself[(long)nclamp * 128 + c];
  }
  __syncthreads();

  const int lane = tid & 31;
  const int wave = tid >> 5;
  const int col0 = wave * 16;          // 8 waves cover 128 output columns
  const int rowA = lane & 15;          // A: lanes 0-15 rows M, lanes 16-31 same rows
  const int kofs = (lane >> 4) * 2;    // A: upper lane half holds K+2,K+3
  const int colB = lane & 15;          // B: column within tile

  v8f acc = {};

  // mean @ Wl : K = 0..127 in steps of 4
  for (int k = 0; k < 128; k += 4) {
    v2f a, b;
    a.x = lmean[rowA * LDS_STRIDE + k + kofs];
    a.y = lmean[rowA * LDS_STRIDE + k + kofs + 1];
    b.x = Wl[(long)(k + kofs) * 128 + col0 + colB];
    b.y = Wl[(long)(k + kofs + 1) * 128 + col0 + colB];
    acc = __builtin_amdgcn_wmma_f32_16x16x4_f32(
        /*neg_a=*/false, a, /*neg_b=*/false, b,
        /*c_mod=*/(short)0, acc, /*reuse_a=*/false, /*reuse_b=*/false);
  }
  // hself @ Wr : K = 0..127
  for (int k = 0; k < 128; k += 4) {
    v2f a, b;
    a.x = lself[rowA * LDS_STRIDE + k + kofs];
    a.y = lself[rowA * LDS_STRIDE + k + kofs + 1];
    b.x = Wr[(long)(k + kofs) * 128 + col0 + colB];
    b.y = Wr[(long)(k + kofs + 1) * 128 + col0 + colB];
    acc = __builtin_amdgcn_wmma_f32_16x16x4_f32(
        false, a, false, b, (short)0, acc, false, false);
  }

  // C/D layout: VGPR i -> M = i (lanes 0-15), M = i+8 (lanes 16-31); N = lane&15
  const int ocol  = col0 + (lane & 15);
  const float bias = bl[ocol];
  const int rbase = row0 + (lane >> 4) * 8;
  for (int i = 0; i < 8; ++i) {
    int node = rbase + i;
    if (node < nNodes) {
      float v = acc[i] + bias;
      if (doRelu) v = fmaxf(v, 0.0f);
      hout[(long)node * 128 + ocol] = v;
    }
  }
}

// ---------------------------------------------------------------------------
// Edge scoring: one wave per edge.
//   y[e] = sigmoid( h2[src].We[0:128] + h2[tgt].We[128:256] + b_e )
// Lane handles 4 elements of each half; wave32 xor-shuffle reduction.
// ---------------------------------------------------------------------------
__global__ void gs_edge_score_kernel(const float* __restrict__ h2,
                                     const int* __restrict__ src,
                                     const int* __restrict__ tgt,
                                     const float* __restrict__ We,
                                     const float* __restrict__ be,
                                     float* __restrict__ out,
                                     int nEdges) {
  int lane = threadIdx.x & 31;
  int e = blockIdx.x * (blockDim.x >> 5) + (threadIdx.x >> 5);
  if (e >= nEdges) return;
  int s = src[e];
  int t = tgt[e];
  const float4 vs = *(const float4*)(h2 + (long)s * 128 + lane * 4);
  const float4 vt = *(const float4*)(h2 + (long)t * 128 + lane * 4);
  const float4 ws = *(const float4*)(We + lane * 4);
  const float4 wt = *(const float4*)(We + 128 + lane * 4);
  float p = vs.x * ws.x + vs.y * ws.y + vs.z * ws.z + vs.w * ws.w
          + vt.x * wt.x + vt.y * wt.y + vt.z * wt.z + vt.w * wt.w;
  for (int off = 16; off > 0; off >>= 1)
    p += __shfl_xor(p, off, 32);
  if (lane == 0) out[e] = 1.0f / (1.0f + __expf(-(p + be[0])));
}

// ---------------------------------------------------------------------------
extern "C" void kernel_launch(void* const* d_in, const int* in_sizes, int n_in,
                              void* d_out, int out_size, void* d_ws, size_t ws_size,
                              hipStream_t stream) {
  const float* x    = (const float*)d_in[0];
  const int*   ei   = (const int*)d_in[1];
  const float* Wl1  = (const float*)d_in[2];
  const float* bl1  = (const float*)d_in[3];
  const float* Wr1  = (const float*)d_in[4];
  const float* Wl2  = (const float*)d_in[5];
  const float* bl2  = (const float*)d_in[6];
  const float* Wr2  = (const float*)d_in[7];
  const float* We   = (const float*)d_in[8];
  const float* be   = (const float*)d_in[9];

  const int N = in_sizes[0] / 128;     // 50000
  const int E = in_sizes[1] / 2;       // 800000
  const int* src = ei;
  const int* tgt = ei + E;

  // Workspace layout (floats): agg[N*128] | cnt[N] | h1[N*128] | h2[N*128]
  float* agg = (float*)d_ws;
  float* cnt = agg + (long)N * 128;
  float* h1  = cnt + N;
  float* h2  = h1 + (long)N * 128;

  const long zn = (long)N * 128 + N;   // agg + cnt are contiguous
  const int zeroBlocks = 2048;
  const int scatBlocks = (E + 7) / 8;  // 8 edges (waves) per 256-thread block
  const int gemmBlocks = (N + 15) / 16;

  // ---- Layer 1: h1 = relu(mean(x_nbrs) @ Wl1 + bl1 + x @ Wr1) ----
  gs_zero_kernel<<<zeroBlocks, 256, 0, stream>>>(agg, zn);
  gs_scatter_kernel<<<scatBlocks, 256, 0, stream>>>(x, src, tgt, agg, cnt, E);
  gs_sage_wmma_kernel<<<gemmBlocks, 256, 0, stream>>>(agg, cnt, x, Wl1, bl1, Wr1,
                                                      h1, /*relu=*/1, N);

  // ---- Layer 2: h2 = mean(h1_nbrs) @ Wl2 + bl2 + h1 @ Wr2 ----
  gs_zero_kernel<<<zeroBlocks, 256, 0, stream>>>(agg, zn);
  gs_scatter_kernel<<<scatBlocks, 256, 0, stream>>>(h1, src, tgt, agg, cnt, E);
  gs_sage_wmma_kernel<<<gemmBlocks, 256, 0, stream>>>(agg, cnt, h1, Wl2, bl2, Wr2,
                                                      h2, /*relu=*/0, N);

  // ---- Edge scores ----
  gs_edge_score_kernel<<<scatBlocks, 256, 0, stream>>>(h2, src, tgt, We, be,
                                                       (float*)d_out, E);
}